// Myloss_19662360281688
// MI455X (gfx1250) — compile-verified
//
#include <hip/hip_runtime.h>
#include <math.h>

typedef __attribute__((ext_vector_type(2))) float v2f;
typedef __attribute__((ext_vector_type(4))) float v4f;
typedef __attribute__((ext_vector_type(8))) float v8f;

#define NBLK  2048   // blocks per array
#define BLOCK 256    // 8 waves (wave32)
#define WAVES (BLOCK / 32)

// ---------------------------------------------------------------------------
// Stage 1: streaming sum-of-exp over both arrays.
// Grid = 2*NBLK blocks; first NBLK handle y_head, second NBLK handle y.
// Accumulation rides the matrix pipe: V_WMMA_F32_16X16X4_F32 with B == 1.0
// gives D[m][n] += sum_k A[m][k]  (exact f32 MACs), co-executing with the
// v_exp_f32 VALU stream. EXEC is all-ones throughout the WMMA loop because
// the iteration count divides exactly (N = 2^25).
// ---------------------------------------------------------------------------
__global__ __launch_bounds__(BLOCK)
void sumexp_kernel(const float* __restrict__ a,
                   const float* __restrict__ b,
                   float* __restrict__ partials,
                   int n4_per_array,   // N/4 float4 elements per array
                   int n_per_array)    // N
{
    const int arr = (blockIdx.x >= NBLK) ? 1 : 0;
    const int blk = arr ? (int)blockIdx.x - NBLK : (int)blockIdx.x;
    const float* __restrict__ src = arr ? b : a;
    const v4f*   __restrict__ src4 = (const v4f*)src;

    const int tid    = blk * BLOCK + (int)threadIdx.x;  // thread id within this array
    const int stride = NBLK * BLOCK;                    // threads per array
    const int iters  = n4_per_array / stride;           // 2^23 / 2^19 = 16, exact

    v8f c = {0.f, 0.f, 0.f, 0.f, 0.f, 0.f, 0.f, 0.f};
    const v2f ones = {1.0f, 1.0f};

    int idx = tid;
    for (int k = 0; k < iters; ++k) {
        v4f v = __builtin_nontemporal_load(&src4[idx]);  // read-once: NT, don't pollute L2
        idx += stride;
        v2f a01 = { __expf(v.x), __expf(v.y) };
        v2f a23 = { __expf(v.z), __expf(v.w) };
        // D = A * ones + C  -> each D slot accumulates a 4-wide row-sum of A
        c = __builtin_amdgcn_wmma_f32_16x16x4_f32(false, a01, false, ones,
                                                  (short)0, c, false, false);
        c = __builtin_amdgcn_wmma_f32_16x16x4_f32(false, a23, false, ones,
                                                  (short)0, c, false, false);
    }

    // Scalar tail for N not divisible by 4*stride (empty for N = 2^25).
    float extra = 0.f;
    for (int i = iters * stride * 4 + tid; i < n_per_array; i += stride)
        extra += __expf(src[i]);
    #pragma unroll
    for (int off = 16; off >= 1; off >>= 1)
        extra += __shfl_xor(extra, off, 32);

    // Every D column is identical: lane l<16 holds rows 0..7, lane l>=16 rows 8..15.
    float s = c[0] + c[1] + c[2] + c[3] + c[4] + c[5] + c[6] + c[7];
    float wave_total = __shfl(s, 0, 32) + __shfl(s, 16, 32) + extra;

    __shared__ float lds[WAVES];
    const int lane = (int)threadIdx.x & 31;
    const int wv   = (int)threadIdx.x >> 5;
    if (lane == 0) lds[wv] = wave_total;
    __syncthreads();
    if (threadIdx.x == 0) {
        float t = 0.f;
        #pragma unroll
        for (int i = 0; i < WAVES; ++i) t += lds[i];   // fixed order -> deterministic
        partials[arr * NBLK + blk] = t;
    }
}

// ---------------------------------------------------------------------------
// Stage 2: single-block deterministic reduction of the 2*NBLK partials and
// closed-form evaluation of the loss. p_head is computed factor-wise
// (exp(x)/Z per factor) to avoid Z^5 overflow; log(p_true) in log form.
// ---------------------------------------------------------------------------
__global__ __launch_bounds__(256)
void finalize_kernel(const float* __restrict__ yh,
                     const float* __restrict__ yt,
                     const int*   __restrict__ kptr,
                     const float* __restrict__ partials,
                     float* __restrict__ out)
{
    __shared__ float lds[256];
    __shared__ float zsh[2];

    for (int arr = 0; arr < 2; ++arr) {
        float t = 0.f;
        #pragma unroll
        for (int j = 0; j < NBLK / 256; ++j)
            t += partials[arr * NBLK + (int)threadIdx.x + j * 256];  // fixed order
        lds[threadIdx.x] = t;
        __syncthreads();
        if (threadIdx.x == 0) {
            float z = 0.f;
            for (int i = 0; i < 256; ++i) z += lds[i];               // fixed order
            zsh[arr] = z;
        }
        __syncthreads();
    }

    if (threadIdx.x == 0) {
        const int K = kptr[0];
        float fact = 1.f;
        for (int i = 2; i <= K; ++i) fact *= (float)i;

        const float Zh = zsh[0];
        const float Zt = zsh[1];

        float p_head = 1.f;   // prod exp(yh[t]) / Zh, factor-wise (no Z^K overflow)
        float sK     = 0.f;   // sum yt[0..K)
        for (int t = 0; t < K; ++t) {
            p_head *= __expf(yh[t]) / Zh;
            sK     += yt[t];
        }
        const float log_p_true = sK - (float)K * __logf(Zt);

        out[0] = -(fact * p_head * log_p_true);
    }
}

// ---------------------------------------------------------------------------
extern "C" void kernel_launch(void* const* d_in, const int* in_sizes, int n_in,
                              void* d_out, int out_size, void* d_ws, size_t ws_size,
                              hipStream_t stream)
{
    const float* y_head = (const float*)d_in[0];
    const float* y_true = (const float*)d_in[1];
    const int*   kptr   = (const int*)d_in[2];
    float*       out    = (float*)d_out;
    float*       parts  = (float*)d_ws;   // 2*NBLK floats = 16 KB, rewritten each call

    const int n  = in_sizes[0];
    const int n4 = n / 4;

    sumexp_kernel<<<2 * NBLK, BLOCK, 0, stream>>>(y_head, y_true, parts, n4, n);
    finalize_kernel<<<1, 256, 0, stream>>>(y_head, y_true, kptr, parts, out);
}